// GraphConv_50276887167203
// MI455X (gfx1250) — compile-verified
//
#include <hip/hip_runtime.h>

#define N_NODES 100000
#define N_EDGES 1600000
#define F 128          // in_feats == out_feats == 128
#define ROW_TILES (N_NODES / 16)   // 6250, exact

typedef __attribute__((ext_vector_type(2))) float v2f;
typedef __attribute__((ext_vector_type(8))) float v8f;

#if __has_builtin(__builtin_amdgcn_wmma_f32_16x16x4_f32)
#define HAVE_WMMA_F32 1
#else
#define HAVE_WMMA_F32 0
#endif

// ---------------- zero workspace (agg + deg) ----------------
__global__ void zero_ws_kernel(float4* __restrict__ p, long n4) {
    long i = (long)blockIdx.x * blockDim.x + threadIdx.x;
    long stride = (long)gridDim.x * blockDim.x;
    float4 z = make_float4(0.f, 0.f, 0.f, 0.f);
    for (; i < n4; i += stride) p[i] = z;
}

// ---------------- edge scatter: wave-per-edge ----------------
// Each wave (32 lanes) handles one edge: lane moves a float4 (512B/wave,
// coalesced) gathered from feat[src] and atomically adds into agg[dst].
// agg (51.2MB) and feat (51.2MB) are L2-resident (192MB L2), so the f32
// atomics retire at the L2 atomic units.
__global__ __launch_bounds__(256) void edge_scatter_kernel(
        const float* __restrict__ feat,
        const int* __restrict__ src,
        const int* __restrict__ dst,
        float* __restrict__ agg,
        float* __restrict__ deg) {
    int wave = blockIdx.x * (blockDim.x >> 5) + (threadIdx.x >> 5);
    int lane = threadIdx.x & 31;
    if (wave >= N_EDGES) return;
    int s = src[wave];
    int d = dst[wave];
    const float4* fs = (const float4*)(feat + (long)s * F);
    float4 v = fs[lane];
    float* ad = agg + (long)d * F + lane * 4;
    atomicAdd(ad + 0, v.x);
    atomicAdd(ad + 1, v.y);
    atomicAdd(ad + 2, v.z);
    atomicAdd(ad + 3, v.w);
    if (lane == 0) atomicAdd(deg + d, 1.0f);
}

// ---------------- GEMM (agg @ weight) + norm + bias ----------------
// One wave per 16-row tile; 8 waves (256 thr) per workgroup. Weight (64KB)
// staged in LDS once per workgroup. fp32 WMMA 16x16x4, K=128 -> 32 k-steps,
// 8 N-tiles -> 256 v_wmma per wave. Epilogue fused: deg^-1/2 clamp + bias.
__global__ __launch_bounds__(256) void gemm_norm_kernel(
        const float* __restrict__ agg,
        const float* __restrict__ weight,
        const float* __restrict__ bias,
        const float* __restrict__ deg,
        float* __restrict__ out) {
    __shared__ float wlds[F * F];   // 64 KB: weight[k][n], row-major K x N

    const int tid = threadIdx.x;
    {
        const float4* wsrc = (const float4*)weight;
        float4* wdst = (float4*)wlds;
        for (int i = tid; i < (F * F) / 4; i += blockDim.x) wdst[i] = wsrc[i];
    }
    __syncthreads();

    const int waveId = tid >> 5;
    const int lane   = tid & 31;
    const int tile   = blockIdx.x * 8 + waveId;
    if (tile >= ROW_TILES) return;      // wave-uniform: EXEC stays all-1s
    const int m0   = tile * 16;
    const int half = lane >> 4;         // 0: lanes 0-15, 1: lanes 16-31
    const int l    = lane & 15;

#if HAVE_WMMA_F32
    v8f acc[8];
#pragma unroll
    for (int n = 0; n < 8; ++n) acc[n] = (v8f)0.0f;

    // A-fragment row for this lane (ISA 7.12.2: A 16x4 f32 layout)
    const float* arow = agg + (long)(m0 + l) * F;

    for (int k0 = 0; k0 < F; k0 += 4) {
        const int ka = k0 + 2 * half;   // lanes 0-15: k0,k0+1; 16-31: k0+2,k0+3
        v2f a;
        a.x = arow[ka];
        a.y = arow[ka + 1];
#pragma unroll
        for (int n = 0; n < 8; ++n) {
            const int col = n * 16 + l;
            v2f b;                       // B 4x16: rows striped over lane halves
            b.x = wlds[ka * F + col];
            b.y = wlds[(ka + 1) * F + col];
            acc[n] = __builtin_amdgcn_wmma_f32_16x16x4_f32(
                false, a, false, b, (short)0, acc[n], false, false);
        }
    }

    // Epilogue: D layout -> lane half selects M block, vgpr v = row-in-half,
    // column = n*16 + (lane&15)
#pragma unroll
    for (int v = 0; v < 8; ++v) {
        const int row = m0 + half * 8 + v;
        float dg = deg[row];
        dg = dg < 1.0f ? 1.0f : dg;
        const float norm = 1.0f / sqrtf(dg);
        float* orow = out + (long)row * F;
#pragma unroll
        for (int n = 0; n < 8; ++n) {
            const int col = n * 16 + l;
            orow[col] = acc[n][v] * norm + bias[col];
        }
    }
#else
    // Scalar fallback (should not be taken on gfx1250): each lane computes
    // rows of its tile at columns lane, lane+32, lane+64, lane+96.
    for (int r = 0; r < 16; ++r) {
        const int row = m0 + r;
        float dg = deg[row];
        dg = dg < 1.0f ? 1.0f : dg;
        const float norm = 1.0f / sqrtf(dg);
        const float* arow = agg + (long)row * F;
        float* orow = out + (long)row * F;
        for (int c = lane; c < F; c += 32) {
            float s = 0.0f;
            for (int k = 0; k < F; ++k) s += arow[k] * wlds[k * F + c];
            orow[c] = s * norm + bias[c];
        }
    }
#endif
}

extern "C" void kernel_launch(void* const* d_in, const int* in_sizes, int n_in,
                              void* d_out, int out_size, void* d_ws, size_t ws_size,
                              hipStream_t stream) {
    (void)in_sizes; (void)n_in; (void)out_size; (void)ws_size;
    const float* feat   = (const float*)d_in[0];
    const int*   src    = (const int*)d_in[1];
    const int*   dst    = (const int*)d_in[2];
    const float* weight = (const float*)d_in[3];
    const float* bias   = (const float*)d_in[4];
    float* out = (float*)d_out;

    float* agg = (float*)d_ws;                       // N_NODES * F floats (51.2 MB)
    float* deg = agg + (size_t)N_NODES * F;          // N_NODES floats

    const long n_zero_floats = (long)N_NODES * F + N_NODES;   // 12,900,000 (div by 4)
    zero_ws_kernel<<<2048, 256, 0, stream>>>((float4*)d_ws, n_zero_floats / 4);

    edge_scatter_kernel<<<N_EDGES / 8, 256, 0, stream>>>(feat, src, dst, agg, deg);

    gemm_norm_kernel<<<(ROW_TILES + 7) / 8, 256, 0, stream>>>(agg, weight, bias, deg, out);
}